// RotaryEmbedding_35433480192112
// MI455X (gfx1250) — compile-verified
//
#include <hip/hip_runtime.h>
#include <math.h>

// Reference constants: DIM=64, ROTATE=4 (first 4 pairs = first 8 floats of
// each 64-float row are rotated; rest pass through).
#define DIM        64
#define ROTATE_N   4
#define QUADS_ROW  (DIM / 4)          // 16 float4s per row
#define ROT_QUADS  (2 * ROTATE_N / 4) // 2 float4s per row are rotated

// Native clang vector type: accepted by __builtin_nontemporal_load/store.
typedef __attribute__((ext_vector_type(4))) float v4f;

// ---------------------------------------------------------------------------
// Setup kernel: direction = sigmoid(dparam[0])*2-1, then
// cs[0..3] = cos(direction*thetas[k]), cs[4..7] = sin(direction*thetas[k]).
// Runs once on 1 block; cost is noise next to the 256 MiB stream.
// ---------------------------------------------------------------------------
__global__ void rope_setup_cs(const float* __restrict__ dparam,
                              const float* __restrict__ thetas,
                              float* __restrict__ cs) {
  int t = threadIdx.x;
  if (t < ROTATE_N) {
    float d   = dparam[0];
    float dir = 1.0f / (1.0f + expf(-d)) * 2.0f - 1.0f;
    float th  = dir * thetas[t];           // TSCALE == 1.0
    cs[t]            = cosf(th);
    cs[t + ROTATE_N] = sinf(th);
  }
}

// ---------------------------------------------------------------------------
// Streaming kernel: grid-stride over 128-bit elements.
// blockDim.x == 256 (multiple of 16) and stride = gridDim*256 (multiple of
// 16), so a thread's quad-in-row index q = threadIdx.x & 15 is LOOP-INVARIANT:
// the rotate predicate and its cos/sin coefficients are hoisted out of the
// loop entirely. Per-iteration body: prefetch, NT b128 load, (masked 8-FMA
// rotate for q<2 lanes), NT b128 store.
// ---------------------------------------------------------------------------
__global__ __launch_bounds__(256) void rope_stream_kernel(
    const v4f* __restrict__ x, v4f* __restrict__ out,
    const float* __restrict__ cs, long long n4) {
  long long i      = (long long)blockIdx.x * blockDim.x + threadIdx.x;
  long long stride = (long long)gridDim.x * blockDim.x;

  // Loop-invariant: requires blockDim.x % 16 == 0 (true: 256).
  const int  q      = (int)(threadIdx.x & (QUADS_ROW - 1));
  const bool rotate = (q < ROT_QUADS);

  // Uniform address -> scalar loads; 8 floats, L2-resident.
  // q==0: pairs (0,1) -> (cs[0],cs[4]),(cs[1],cs[5])
  // q==1: pairs (2,3) -> (cs[2],cs[6]),(cs[3],cs[7])
  const float cA = q ? cs[2] : cs[0];
  const float cB = q ? cs[3] : cs[1];
  const float sA = q ? cs[6] : cs[4];
  const float sB = q ? cs[7] : cs[5];

  for (; i < n4; i += stride) {
    long long inext = i + stride;
    if (inext < n4) {
      __builtin_prefetch(&x[inext], 0, 0);   // global_prefetch_b8 (speculative)
    }

    v4f v = __builtin_nontemporal_load(&x[i]);   // global_load_b128 th:NT

    if (rotate) {
      float xi0 = v.x, xj0 = v.y;
      float xi1 = v.z, xj1 = v.w;
      v.x = fmaf(xi0, cA,  xj0 * sA);   // new_i =  xi*c + xj*s
      v.y = fmaf(xj0, cA, -xi0 * sA);   // new_j = -xi*s + xj*c
      v.z = fmaf(xi1, cB,  xj1 * sB);
      v.w = fmaf(xj1, cB, -xi1 * sB);
    }

    __builtin_nontemporal_store(v, &out[i]);     // global_store_b128 th:NT
  }
}

// ---------------------------------------------------------------------------
// Launch: inputs are (x, dparam, thetas) per setup_inputs() order.
// ---------------------------------------------------------------------------
extern "C" void kernel_launch(void* const* d_in, const int* in_sizes, int n_in,
                              void* d_out, int out_size, void* d_ws, size_t ws_size,
                              hipStream_t stream) {
  const float* x      = (const float*)d_in[0];
  const float* dparam = (const float*)d_in[1];
  const float* thetas = (const float*)d_in[2];
  float*       cs     = (float*)d_ws;   // 8 floats of scratch

  rope_setup_cs<<<1, 32, 0, stream>>>(dparam, thetas, cs);

  long long n  = (long long)in_sizes[0];   // 4*16*8192*64 = 33,554,432 floats
  long long n4 = n >> 2;                   // 8,388,608 float4s

  const int block     = 256;               // 8 wave32s; multiple of 16 (required)
  const int iters     = 4;                 // grid-stride depth (enables prefetch)
  long long per_block = (long long)block * iters;
  int grid = (int)((n4 + per_block - 1) / per_block);   // 8192 blocks

  rope_stream_kernel<<<grid, block, 0, stream>>>(
      (const v4f*)x, (v4f*)d_out, cs, n4);
}